// BipartiteGCN_47751446397296
// MI455X (gfx1250) — compile-verified
//
#include <hip/hip_runtime.h>
#include <hip/hip_bf16.h>

typedef __attribute__((ext_vector_type(2))) float v2f;
typedef __attribute__((ext_vector_type(4))) float v4f;
typedef __attribute__((ext_vector_type(8))) float v8f;

#define N_READ   200000
#define N_INTRON 50000
#define N_EDGES  2000000

// ---------------------------------------------------------------- utilities
__global__ void k_zero_u32(unsigned* __restrict__ p, int n) {
  int i = blockIdx.x * blockDim.x + threadIdx.x;
  if (i < n) p[i] = 0u;
}

__global__ void k_copy_u32(unsigned* __restrict__ d, const unsigned* __restrict__ s, int n) {
  int i = blockIdx.x * blockDim.x + threadIdx.x;
  if (i < n) d[i] = s[i];
}

// ---------------------------------------------------------------- degrees
__global__ void k_degree(const int* __restrict__ src, const int* __restrict__ dst,
                         unsigned* __restrict__ deg_r, unsigned* __restrict__ deg_i, int ne) {
  int e = blockIdx.x * blockDim.x + threadIdx.x;
  if (e >= ne) return;
  atomicAdd(&deg_r[src[e]], 1u);
  atomicAdd(&deg_i[dst[e]], 1u);
}

__global__ void k_rsq(const unsigned* __restrict__ deg, float* __restrict__ rsq, int n) {
  int i = blockIdx.x * blockDim.x + threadIdx.x;
  if (i < n) rsq[i] = rsqrtf(fmaxf((float)deg[i], 1.0f));
}

// ---------------------------------------------------------------- exclusive scan (3 kernels)
__global__ void k_scan_block(const unsigned* __restrict__ in, unsigned* __restrict__ out,
                             unsigned* __restrict__ bsums, int n) {
  __shared__ unsigned sh[256];
  int tid = threadIdx.x;
  int g = blockIdx.x * 256 + tid;
  unsigned v = (g < n) ? in[g] : 0u;
  sh[tid] = v;
  __syncthreads();
  for (int off = 1; off < 256; off <<= 1) {
    unsigned t = (tid >= off) ? sh[tid - off] : 0u;
    __syncthreads();
    sh[tid] += t;
    __syncthreads();
  }
  if (g < n) out[g] = sh[tid] - v;          // exclusive
  if (tid == 255) bsums[blockIdx.x] = sh[255];
}

__global__ void k_scan_top(unsigned* __restrict__ bsums, int nb) {  // one block, 1024 threads
  __shared__ unsigned sh[1024];
  int tid = threadIdx.x;
  unsigned v = (tid < nb) ? bsums[tid] : 0u;
  sh[tid] = v;
  __syncthreads();
  for (int off = 1; off < 1024; off <<= 1) {
    unsigned t = (tid >= off) ? sh[tid - off] : 0u;
    __syncthreads();
    sh[tid] += t;
    __syncthreads();
  }
  if (tid < nb) bsums[tid] = sh[tid] - v;   // exclusive
}

__global__ void k_scan_add(unsigned* __restrict__ out, const unsigned* __restrict__ bsums,
                           int n, unsigned total) {
  int g = blockIdx.x * 256 + threadIdx.x;
  if (g < n)       out[g] += bsums[blockIdx.x];
  else if (g == n) out[g] = total;          // offs[n] = n_edges
}

// ---------------------------------------------------------------- CSR bucket fill (both directions)
__global__ void k_csr_fill(const int* __restrict__ src, const int* __restrict__ dst,
                           unsigned* __restrict__ cur_r, unsigned* __restrict__ cur_i,
                           int* __restrict__ nbr_r, int* __restrict__ nbr_i, int ne) {
  int e = blockIdx.x * blockDim.x + threadIdx.x;
  if (e >= ne) return;
  int s = src[e], d = dst[e];
  unsigned p = atomicAdd(&cur_r[s], 1u);  nbr_r[p] = d;   // read-keyed list of intron neighbors
  unsigned q = atomicAdd(&cur_i[d], 1u);  nbr_i[q] = s;   // intron-keyed list of read neighbors
}

// ---------------------------------------------------------------- pull aggregation (no float atomics)
// One wave per dst node; lanes cooperatively load 32 neighbor ids, broadcast
// via shuffle, then do coalesced b128/b64 row reads. Source-degree scaling was
// folded into the producing layer's epilogue, so this is a pure sum.
__global__ void k_pull128(const float* __restrict__ feat, const unsigned* __restrict__ offs,
                          const int* __restrict__ nbrs, float* __restrict__ agg, int ndst) {
  int wave = (int)((blockIdx.x * blockDim.x + threadIdx.x) >> 5);
  int lane = (int)(threadIdx.x & 31);
  if (wave >= ndst) return;
  unsigned s0 = offs[wave], s1 = offs[wave + 1];
  v4f acc = {0.f, 0.f, 0.f, 0.f};
  for (unsigned base = s0; base < s1; base += 32u) {
    int my = (base + (unsigned)lane < s1) ? nbrs[base + lane] : 0;
    unsigned rem = s1 - base;
    int cnt = rem < 32u ? (int)rem : 32;
    for (int j = 0; j < cnt; j++) {
      int nb = __shfl(my, j);
      acc += *(const v4f*)(feat + (size_t)nb * 128 + lane * 4);
    }
  }
  *(v4f*)(agg + (size_t)wave * 128 + lane * 4) = acc;
}

__global__ void k_pull64(const float* __restrict__ feat, const unsigned* __restrict__ offs,
                         const int* __restrict__ nbrs, float* __restrict__ agg, int ndst) {
  int wave = (int)((blockIdx.x * blockDim.x + threadIdx.x) >> 5);
  int lane = (int)(threadIdx.x & 31);
  if (wave >= ndst) return;
  unsigned s0 = offs[wave], s1 = offs[wave + 1];
  v2f acc = {0.f, 0.f};
  for (unsigned base = s0; base < s1; base += 32u) {
    int my = (base + (unsigned)lane < s1) ? nbrs[base + lane] : 0;
    unsigned rem = s1 - base;
    int cnt = rem < 32u ? (int)rem : 32;
    for (int j = 0; j < cnt; j++) {
      int nb = __shfl(my, j);
      acc += *(const v2f*)(feat + (size_t)nb * 64 + lane * 2);
    }
  }
  *(v2f*)(agg + (size_t)wave * 64 + lane * 2) = acc;
}

// Layer 0: pull raw 10-dim read features, scale by rsqrt(out_deg_read),
// write zero-padded to lda=16 (pads the K dim for the 16x16x4 WMMA loop).
__global__ void k_pull10(const float* __restrict__ feat, const float* __restrict__ rsq_src,
                         const unsigned* __restrict__ offs, const int* __restrict__ nbrs,
                         float* __restrict__ agg, int ndst) {
  int wave = (int)((blockIdx.x * blockDim.x + threadIdx.x) >> 5);
  int lane = (int)(threadIdx.x & 31);
  if (wave >= ndst) return;
  unsigned s0 = offs[wave], s1 = offs[wave + 1];
  float acc = 0.0f;
  for (unsigned base = s0; base < s1; base += 32u) {
    int my = (base + (unsigned)lane < s1) ? nbrs[base + lane] : 0;
    unsigned rem = s1 - base;
    int cnt = rem < 32u ? (int)rem : 32;
    for (int j = 0; j < cnt; j++) {
      int nb = __shfl(my, j);
      float sc = rsq_src[nb];
      float v = (lane < 10) ? feat[(size_t)nb * 10 + lane] : 0.0f;
      acc += v * sc;
    }
  }
  if (lane < 16) agg[(size_t)wave * 16 + lane] = acc;   // cols 10..15 are zero
}

// ---------------------------------------------------------------- fused WMMA GEMM + GCN epilogue
// out[m,n] = relu( gate * ( rsq_dst[m] * sum_k A[m,k]*W[k,n] + bias[n] ) ) * (FOLD ? rsq_next[m] : 1)
//
// All shape params are compile-time: no guards or branches anywhere. W is
// staged once per block into LDS in WMMA-fragment-swizzled order (zero-padded
// for K >= FIN), so each B fragment is ONE conflict-free ds_load_b64 per lane
// (lane*8B covers all 64 banks exactly once; pairs merge to ds_load_2addr_b64).
// A fragments are single global_load_b64 per lane. One wave owns a 16-row tile
// across all NT 16-column tiles, reusing the A fragment NT times per k-step.
// f32 WMMA fragment layout per ISA: lanes 0-15 carry K=k0,k0+1 ; lanes 16-31 K=k0+2,k0+3.
template <int NT, int KPAD, int FIN, bool FOLD>
__global__ void k_gemm_wmma(const float* __restrict__ A,
                            const float* __restrict__ Wg,
                            const float* __restrict__ bias,
                            const float* __restrict__ rsq_dst,
                            const float* __restrict__ rsq_next,
                            const float* __restrict__ atts, int layer,
                            float* __restrict__ out, int nrows) {
  constexpr int FOUT = NT * 16;
  __shared__ float lw[KPAD * FOUT];   // swizzled: [(k0/4)*NT + t][lane][2]

  // cooperative fragment-swizzled staging of W (with zero K-padding)
  for (int o = (int)threadIdx.x; o < (KPAD / 4) * NT * 32; o += (int)blockDim.x) {
    int kb = o / (NT * 32);
    int r  = o % (NT * 32);
    int t  = r >> 5;
    int ln = r & 31;
    int k  = kb * 4 + (ln >> 4) * 2;
    int n  = t * 16 + (ln & 15);
    float x = (k     < FIN) ? Wg[(size_t)k * FOUT + n]       : 0.0f;
    float y = (k + 1 < FIN) ? Wg[(size_t)(k + 1) * FOUT + n] : 0.0f;
    lw[o * 2]     = x;
    lw[o * 2 + 1] = y;
  }
  __syncthreads();

  int wave = (int)((blockIdx.x * blockDim.x + threadIdx.x) >> 5);
  int lane = (int)(threadIdx.x & 31);
  int m0 = wave * 16;
  if (m0 + 16 > nrows) m0 = nrows - 16;    // clamp (redundant recompute of last
                                           // tile, identical values): keeps EXEC
                                           // all-1s for WMMA and all waves at the barrier
  int hb = lane >> 4;
  int l16 = lane & 15;
  float gate = 1.0f / (1.0f + __expf(-atts[layer]));

  v8f acc[NT];
  const v8f vz = {0.f, 0.f, 0.f, 0.f, 0.f, 0.f, 0.f, 0.f};
#pragma unroll
  for (int t = 0; t < NT; t++) acc[t] = vz;

  const float* arow = A + (size_t)(m0 + l16) * KPAD;
#pragma unroll 2
  for (int k0 = 0; k0 < KPAD; k0 += 4) {
    v2f af = *(const v2f*)(arow + k0 + hb * 2);                // global_load_b64
    const float* wp = lw + ((k0 >> 2) * NT) * 64 + lane * 2;
#pragma unroll
    for (int t = 0; t < NT; t++) {
      v2f bf = *(const v2f*)(wp + t * 64);                     // ds_load_b64, bank-conflict-free
      acc[t] = __builtin_amdgcn_wmma_f32_16x16x4_f32(
          /*neg_a=*/false, af, /*neg_b=*/false, bf,
          /*c_mod=*/(short)0, acc[t], /*reuse_a=*/false, /*reuse_b=*/false);
    }
  }

  // C/D layout: lanes 0-15 -> M = m0+v, lanes 16-31 -> M = m0+8+v ; N = t*16 + l16
  float rs[8], rn[8];
#pragma unroll
  for (int v = 0; v < 8; v++) {
    int m = m0 + hb * 8 + v;
    rs[v] = rsq_dst[m];
    rn[v] = FOLD ? rsq_next[m] : 1.0f;     // compile-time: no branches, no null check
  }
#pragma unroll
  for (int t = 0; t < NT; t++) {
    int n = t * 16 + l16;
    float bv = bias[n];
#pragma unroll
    for (int v = 0; v < 8; v++) {
      int m = m0 + hb * 8 + v;
      float val = acc[t][v] * rs[v] + bv;
      val = fmaxf(gate * val, 0.0f);
      val *= rn[v];   // fold next layer's rsqrt(out_deg): positive scale commutes with relu
      out[(size_t)m * FOUT + n] = val;
    }
  }
}

// ---------------------------------------------------------------- final FC: [N,128] @ [128,2] + b
__global__ void k_fc(const float* __restrict__ h, const float* __restrict__ w,
                     const float* __restrict__ b, float* __restrict__ out, int n) {
  int wave = (int)((blockIdx.x * blockDim.x + threadIdx.x) >> 5);
  int lane = (int)(threadIdx.x & 31);
  if (wave >= n) return;
  v4f hv = *(const v4f*)(h + (size_t)wave * 128 + lane * 4);
  float a0 = 0.0f, a1 = 0.0f;
#pragma unroll
  for (int q = 0; q < 4; q++) {
    int k = lane * 4 + q;
    a0 += hv[q] * w[k * 2 + 0];
    a1 += hv[q] * w[k * 2 + 1];
  }
  for (int off = 16; off > 0; off >>= 1) {
    a0 += __shfl_xor(a0, off);
    a1 += __shfl_xor(a1, off);
  }
  if (lane == 0) {
    out[(size_t)wave * 2 + 0] = a0 + b[0];
    out[(size_t)wave * 2 + 1] = a1 + b[1];
  }
}

// ---------------------------------------------------------------- launch
static inline int cdiv(int a, int b) { return (a + b - 1) / b; }

extern "C" void kernel_launch(void* const* d_in, const int* in_sizes, int n_in,
                              void* d_out, int out_size, void* d_ws, size_t ws_size,
                              hipStream_t stream) {
  (void)in_sizes; (void)n_in; (void)out_size; (void)ws_size;
  const float* h_read = (const float*)d_in[0];
  const int*   esrc   = (const int*)d_in[1];
  const int*   edst   = (const int*)d_in[2];
  const float* W[6]; const float* B[6];
  for (int i = 0; i < 6; i++) { W[i] = (const float*)d_in[3 + 2 * i]; B[i] = (const float*)d_in[4 + 2 * i]; }
  const float* atts = (const float*)d_in[15];
  const float* fcw  = (const float*)d_in[16];
  const float* fcb  = (const float*)d_in[17];
  float* out = (float*)d_out;

  // workspace carve-out
  char* ws = (char*)d_ws;
  size_t off = 0;
  auto alloc = [&](size_t bytes) -> char* {
    char* p = ws + off;
    off = (off + bytes + 255) & ~(size_t)255;
    return p;
  };
  unsigned* deg_r  = (unsigned*)alloc((size_t)N_READ * 4);
  unsigned* deg_i  = (unsigned*)alloc((size_t)N_INTRON * 4);
  unsigned* offs_r = (unsigned*)alloc((size_t)(N_READ + 1) * 4);
  unsigned* offs_i = (unsigned*)alloc((size_t)(N_INTRON + 1) * 4);
  unsigned* cur_r  = (unsigned*)alloc((size_t)N_READ * 4);
  unsigned* cur_i  = (unsigned*)alloc((size_t)N_INTRON * 4);
  unsigned* bsums  = (unsigned*)alloc(1024 * 4);
  float* rsq_r = (float*)alloc((size_t)N_READ * 4);
  float* rsq_i = (float*)alloc((size_t)N_INTRON * 4);
  int* nbr_r = (int*)alloc((size_t)N_EDGES * 4);     // keyed by read node -> intron ids
  int* nbr_i = (int*)alloc((size_t)N_EDGES * 4);     // keyed by intron node -> read ids
  float* agg_i = (float*)alloc((size_t)N_INTRON * 128 * 4);
  float* agg_r = (float*)alloc((size_t)N_READ * 128 * 4);
  float* h_i   = (float*)alloc((size_t)N_INTRON * 128 * 4);
  float* h_r   = (float*)alloc((size_t)N_READ * 128 * 4);

  const int bR = cdiv(N_READ, 256), bI = cdiv(N_INTRON, 256), bE = cdiv(N_EDGES, 256);

  // ---- graph structure (recomputed every call; counters start at zero)
  k_zero_u32<<<bR, 256, 0, stream>>>(deg_r, N_READ);
  k_zero_u32<<<bI, 256, 0, stream>>>(deg_i, N_INTRON);
  k_degree<<<bE, 256, 0, stream>>>(esrc, edst, deg_r, deg_i, N_EDGES);
  k_rsq<<<bR, 256, 0, stream>>>(deg_r, rsq_r, N_READ);
  k_rsq<<<bI, 256, 0, stream>>>(deg_i, rsq_i, N_INTRON);

  k_scan_block<<<bR, 256, 0, stream>>>(deg_r, offs_r, bsums, N_READ);
  k_scan_top<<<1, 1024, 0, stream>>>(bsums, bR);
  k_scan_add<<<bR, 256, 0, stream>>>(offs_r, bsums, N_READ, (unsigned)N_EDGES);

  k_scan_block<<<bI, 256, 0, stream>>>(deg_i, offs_i, bsums, N_INTRON);
  k_scan_top<<<1, 1024, 0, stream>>>(bsums, bI);
  k_scan_add<<<bI, 256, 0, stream>>>(offs_i, bsums, N_INTRON, (unsigned)N_EDGES);

  k_copy_u32<<<bR, 256, 0, stream>>>(cur_r, offs_r, N_READ);
  k_copy_u32<<<bI, 256, 0, stream>>>(cur_i, offs_i, N_INTRON);
  k_csr_fill<<<bE, 256, 0, stream>>>(esrc, edst, cur_r, cur_i, nbr_r, nbr_i, N_EDGES);

  // wave-per-node grids
  const int pR = N_READ / 8, pI = N_INTRON / 8;                 // pull: 8 waves/block
  const int gR = cdiv((N_READ / 16) * 32, 256);                 // gemm: 16 rows per wave
  const int gI = cdiv((N_INTRON / 16) * 32, 256);

  // L0: read -> intron (fin 10 padded to 16, fout 64). rsq_next folds L1 source scale.
  k_pull10<<<pI, 256, 0, stream>>>(h_read, rsq_r, offs_i, nbr_i, agg_i, N_INTRON);
  k_gemm_wmma<4, 16, 10, true><<<gI, 256, 0, stream>>>(agg_i, W[0], B[0], rsq_i, rsq_i, atts, 0, h_i, N_INTRON);
  // L1: intron -> read (64 -> 128)
  k_pull64<<<pR, 256, 0, stream>>>(h_i, offs_r, nbr_r, agg_r, N_READ);
  k_gemm_wmma<8, 64, 64, true><<<gR, 256, 0, stream>>>(agg_r, W[1], B[1], rsq_r, rsq_r, atts, 1, h_r, N_READ);
  // L2: read -> intron (128 -> 128)
  k_pull128<<<pI, 256, 0, stream>>>(h_r, offs_i, nbr_i, agg_i, N_INTRON);
  k_gemm_wmma<8, 128, 128, true><<<gI, 256, 0, stream>>>(agg_i, W[2], B[2], rsq_i, rsq_i, atts, 2, h_i, N_INTRON);
  // L3: intron -> read
  k_pull128<<<pR, 256, 0, stream>>>(h_i, offs_r, nbr_r, agg_r, N_READ);
  k_gemm_wmma<8, 128, 128, true><<<gR, 256, 0, stream>>>(agg_r, W[3], B[3], rsq_r, rsq_r, atts, 3, h_r, N_READ);
  // L4: read -> intron
  k_pull128<<<pI, 256, 0, stream>>>(h_r, offs_i, nbr_i, agg_i, N_INTRON);
  k_gemm_wmma<8, 128, 128, true><<<gI, 256, 0, stream>>>(agg_i, W[4], B[4], rsq_i, rsq_i, atts, 4, h_i, N_INTRON);
  // L5: intron -> read (no next-layer fold; FC consumes h_r directly)
  k_pull128<<<pR, 256, 0, stream>>>(h_i, offs_r, nbr_r, agg_r, N_READ);
  k_gemm_wmma<8, 128, 128, false><<<gR, 256, 0, stream>>>(agg_r, W[5], B[5], rsq_r, nullptr, atts, 5, h_r, N_READ);

  // final FC: [N_READ,128] @ [128,2] + b
  k_fc<<<N_READ / 8, 256, 0, stream>>>(h_r, fcw, fcb, out, N_READ);
}